// KalmanFilter_5815385719315
// MI455X (gfx1250) — compile-verified
//
#include <hip/hip_runtime.h>

// Kalman filter, L = D = U = 64, T = 8192.
// Single persistent workgroup (1 WGP), all state in LDS (~210 KB of 320 KB),
// FP32 WMMA (v_wmma_f32_16x16x4_f32) for all 64x64x64 matmuls.

constexpr int N  = 64;    // L == D == U
constexpr int TT = 8192;  // sequence length
constexpr int AUGS = 132; // padded row stride of augmented [S | M] buffer
constexpr int NTHREADS = 512; // 16 wave32 -> one 16x16 tile per wave

typedef float v2f __attribute__((ext_vector_type(2)));
typedef float v8f __attribute__((ext_vector_type(8)));

// One 16x16 f32 tile of D = X @ Y (+ Cin), K = 64, via 16x v_wmma_f32_16x16x4_f32.
// X, Y, Cin row-major stride 64 in LDS; D row-major stride `dstride`.
// Fragment layouts per CDNA5 ISA 7.12.2:
//   A 16x4 f32 : lane(0-15) row M=lane, v0=K0,v1=K1 ; lanes 16-31: K2,K3
//   C/D 16x16  : v[i] = row (i + 8*(lane>>4)), col = lane&15
__device__ __forceinline__ void mm64_tile(const float* __restrict__ X,
                                          const float* __restrict__ Y,
                                          const float* __restrict__ Cin,
                                          float* __restrict__ D, int dstride,
                                          int tm16, int tn16, int m, int kh)
{
    v8f acc;
    if (Cin) {
#pragma unroll
        for (int i = 0; i < 8; ++i) acc[i] = Cin[(tm16 + i + 8*kh)*N + tn16 + m];
    } else {
#pragma unroll
        for (int i = 0; i < 8; ++i) acc[i] = 0.0f;
    }
#pragma unroll
    for (int k = 0; k < N; k += 4) {
        const int ka = k + 2*kh;
        v2f a, b;
        a.x = X[(tm16 + m)*N + ka];
        a.y = X[(tm16 + m)*N + ka + 1];
        b.x = Y[ ka     *N + tn16 + m];
        b.y = Y[(ka + 1)*N + tn16 + m];
        acc = __builtin_amdgcn_wmma_f32_16x16x4_f32(
                  false, a, false, b, (short)0, acc, false, false);
    }
#pragma unroll
    for (int i = 0; i < 8; ++i) D[(tm16 + i + 8*kh)*dstride + tn16 + m] = acc[i];
}

__global__ void __launch_bounds__(NTHREADS)
kalman_kernel(const float* __restrict__ inputs,   // [D, T]
              const float* __restrict__ controls, // [U, T]
              const float* __restrict__ Ag,       // [L, L]
              const float* __restrict__ Bg,       // [L, U]
              const float* __restrict__ Cg,       // [D, L]
              const float* __restrict__ Qg,       // [L, L]
              const float* __restrict__ Rg,       // [D, D]
              float* __restrict__ out)            // [L, T]
{
    __shared__ float sA [N*N], sAT[N*N], sB [N*N], sC [N*N], sCT[N*N];
    __shared__ float sQ [N*N], sR [N*N];
    __shared__ float sP [N*N], sPp[N*N], sM [N*N], sT1[N*N];
    __shared__ float sAug[N*AUGS];                 // [S | M] augmented, padded rows
    __shared__ float sRd[N], sZ[N], sZp[N], sInn[N];

    const int tid  = threadIdx.x;
    const int lane = tid & 31;
    const int wave = tid >> 5;
    const int m    = lane & 15;
    const int kh   = lane >> 4;
    const int tm16 = (wave >> 2) * 16;             // 4x4 grid of 16x16 tiles
    const int tn16 = (wave &  3) * 16;

    // ---- setup: stage constants into LDS, build explicit transposes, P = I, z = 0
    for (int i = tid; i < N*N; i += NTHREADS) {
        const int r = i >> 6, c = i & 63;
        const float a = Ag[i], cc = Cg[i];
        sA[i] = a;  sC[i] = cc;
        sAT[c*N + r] = a;  sCT[c*N + r] = cc;
        sB[i] = Bg[i];  sQ[i] = Qg[i];  sR[i] = Rg[i];
        sP[i] = (r == c) ? 1.0f : 0.0f;
    }
    if (tid < N) sZ[tid] = 0.0f;
    __syncthreads();

    for (int t = 0; t < TT; ++t) {
        // ---- S1: z_proj = A z + B u  (waves 0-1)  ||  T1 = A @ P (all 16 waves)
        if (tid < N) {
            float s = 0.0f;
            for (int j = 0; j < N; ++j) {
                s += sA[tid*N + j] * sZ[j];
                s += sB[tid*N + j] * controls[j*TT + t];
            }
            sZp[tid] = s;
        }
        mm64_tile(sA, sP, nullptr, sT1, N, tm16, tn16, m, kh);
        __syncthreads();

        // ---- S2: P_proj = T1 @ A^T + Q
        mm64_tile(sT1, sAT, sQ, sPp, N, tm16, tn16, m, kh);
        __syncthreads();

        // ---- S3: M = C @ P_proj
        mm64_tile(sC, sPp, nullptr, sM, N, tm16, tn16, m, kh);
        __syncthreads();

        // ---- S4: S = M @ C^T + R -> Aug left half; copy M -> Aug right half;
        //          innov = x_t - C z_proj (waves 0-1)
        mm64_tile(sM, sCT, sR, sAug, AUGS, tm16, tn16, m, kh);
        for (int i = tid; i < N*N; i += NTHREADS)
            sAug[(i >> 6)*AUGS + N + (i & 63)] = sM[i];
        if (tid < N) {
            float s = inputs[tid*TT + t];
            for (int j = 0; j < N; ++j) s -= sC[tid*N + j] * sZp[j];
            sInn[tid] = s;
        }

        // ---- S5: Gauss-Jordan solve S X = M  (unscaled variant, 1 barrier/pivot).
        // Pivot column writes are skipped so f-reads race with nothing; the pivot
        // value at pivot time is the final left-diagonal -> save its reciprocal.
        {
            const int gr  = tid >> 3;          // 8 threads per row
            const int gc0 = (tid & 7) * 16;    // 16 columns each
            for (int p = 0; p < N; ++p) {
                __syncthreads();               // also covers S4 completion at p==0
                const float piv = sAug[p*AUGS + p];
                if (tid == 0) sRd[p] = 1.0f / piv;
                if (gr != p) {
                    const float f = sAug[gr*AUGS + p] / piv;
                    const float* prow = &sAug[p *AUGS];
                    float*       rrow = &sAug[gr*AUGS];
#pragma unroll
                    for (int j = 0; j < 16; ++j) {
                        const int c = gc0 + j;
                        const float v = rrow[c] - f * prow[c];
                        if (c != p) rrow[c] = v;
                    }
                }
            }
        }
        __syncthreads();

        // ---- S6: P_new = P_proj - X^T @ M  (neg+scale fused into A-frag load,
        //          accumulator preloaded from P_proj), and z_new + output column.
        {
            v8f acc;
#pragma unroll
            for (int i = 0; i < 8; ++i) acc[i] = sPp[(tm16 + i + 8*kh)*N + tn16 + m];
#pragma unroll
            for (int k = 0; k < N; k += 4) {
                const int ka = k + 2*kh;
                v2f a, b;
                // X^T[row][ka] = Aug[ka][64+row] * rdiag[ka], negated
                a.x = -sAug[ ka     *AUGS + N + tm16 + m] * sRd[ka];
                a.y = -sAug[(ka + 1)*AUGS + N + tm16 + m] * sRd[ka + 1];
                b.x = sM[ ka     *N + tn16 + m];
                b.y = sM[(ka + 1)*N + tn16 + m];
                acc = __builtin_amdgcn_wmma_f32_16x16x4_f32(
                          false, a, false, b, (short)0, acc, false, false);
            }
#pragma unroll
            for (int i = 0; i < 8; ++i) sP[(tm16 + i + 8*kh)*N + tn16 + m] = acc[i];
        }
        if (tid < N) {
            float s = sZp[tid];
            for (int d = 0; d < N; ++d)
                s += sAug[d*AUGS + N + tid] * sRd[d] * sInn[d];
            sZ[tid] = s;
            out[tid*TT + t] = s;              // zs[l, t]
        }
        __syncthreads();
    }
}

extern "C" void kernel_launch(void* const* d_in, const int* in_sizes, int n_in,
                              void* d_out, int out_size, void* d_ws, size_t ws_size,
                              hipStream_t stream) {
    const float* inputs   = (const float*)d_in[0];
    const float* controls = (const float*)d_in[1];
    const float* A        = (const float*)d_in[2];
    const float* B        = (const float*)d_in[3];
    const float* C        = (const float*)d_in[4];
    const float* Q        = (const float*)d_in[5];
    const float* R        = (const float*)d_in[6];
    float* out = (float*)d_out;
    kalman_kernel<<<dim3(1), dim3(NTHREADS), 0, stream>>>(
        inputs, controls, A, B, C, Q, R, out);
}